// DAG_RNN_81965155876934
// MI455X (gfx1250) — compile-verified
//
#include <hip/hip_runtime.h>
#include <math.h>

// ---------------------------------------------------------------------------
// DAG-GRU for MI455X (gfx1250, wave32).
//   B=64, T=512, D=256, U=256, G=3U=768
//   Phase 1 (parallel):  x_proj[t,b,g] = x[b,t,:] @ kernel + bias[0]   (WMMA f32)
//   Phase 2 (serial, per t):
//     h_prev[b,u] = sum_{j<t} cond[b,t,j] * H[j,b,u]                  (VALU, L2-resident)
//     rec = h_prev @ W_rec + bias[1]; GRU gates; write H[t], out[:,t]  (WMMA f32, fused gates)
// ---------------------------------------------------------------------------

typedef __attribute__((ext_vector_type(2))) float v2f;
typedef __attribute__((ext_vector_type(8))) float v8f;

#define BB   64
#define TT   512
#define DD   256
#define UU   256
#define GG   768     // 3*U
#define BU   (BB*UU)         // 16384
#define BG   (BB*GG)         // 49152

// ------------------------------------------------------------------
// Kernel 1: x_proj = x @ Wk + bias0.
//   A = x flattened [M=B*T, K=256]  (row m = (b,t), contiguous)
//   B = Wk [K=256, N=768] row-major
//   Output scattered into xp[T][B][G].
// One 16x16 tile per wave, 4 waves per block. K consumed 4 at a time
// via V_WMMA_F32_16X16X4_F32.
// ------------------------------------------------------------------
__global__ __launch_bounds__(128) void xproj_wmma_kernel(
    const float* __restrict__ x, const float* __restrict__ Wk,
    const float* __restrict__ bias, float* __restrict__ xp)
{
    const int lane = threadIdx.x;          // 0..31
    const int lr   = lane & 15;            // 0..15
    const int hi   = lane >> 4;            // 0 or 1

    const int tile   = blockIdx.x * 4 + threadIdx.y;   // 0..98303
    const int tile_n = tile % 48;                      // 48 N-tiles
    const int tile_m = tile / 48;                      // 2048 M-tiles

    const int m = tile_m * 16 + lr;                    // A-row for this lane
    const int n = tile_n * 16 + lr;                    // B/D column for this lane

    const float* arow = x + (size_t)m * DD;            // x row (b*T+t)*D
    v8f acc = {};

#pragma unroll 8
    for (int kk = 0; kk < DD; kk += 4) {
        const int k0 = kk + 2 * hi;
        // A 16x4 fragment: lane holds A[m][k0], A[m][k0+1]  -> one b64 load
        v2f a = *(const v2f*)(arow + k0);
        // B 4x16 fragment: lane holds Wk[k0][n], Wk[k0+1][n]
        v2f b;
        b.x = Wk[(size_t)k0 * GG + n];
        b.y = Wk[(size_t)(k0 + 1) * GG + n];
        acc = __builtin_amdgcn_wmma_f32_16x16x4_f32(false, a, false, b,
                                                    (short)0, acc, false, false);
    }

    // C/D layout: VGPR e, lane l -> M = e + 8*(l>=16), N = l%16
    const int mbase = tile_m * 16;
#pragma unroll
    for (int e = 0; e < 8; ++e) {
        const int mm = mbase + e + 8 * hi;     // (b,t) flat: b = mm>>9, t = mm&511
        const int b  = mm >> 9;
        const int t  = mm & (TT - 1);
        xp[(size_t)(t * BB + b) * GG + n] = acc[e] + bias[n];   // bias row 0
    }
}

// ------------------------------------------------------------------
// Kernel 2 (per step t): h_prev[b,u] = sum_{j<t} cond[b,t,j]*H[j,b,u]
// H layout [T][B][U]: lanes stride over u -> fully coalesced; H fits in L2.
// ------------------------------------------------------------------
__global__ __launch_bounds__(256) void hprev_kernel(
    const float* __restrict__ cond, const float* __restrict__ H,
    float* __restrict__ hprev, int t)
{
    const int idx = blockIdx.x * 256 + threadIdx.x;    // 0..16383
    const int b = idx >> 8;
    const int u = idx & (UU - 1);

    const float* crow = cond + ((size_t)b * TT + t) * TT;  // cond[b,t,:], wave-uniform
    const float* Hb   = H + (size_t)b * UU + u;

    float acc = 0.0f;
#pragma unroll 4
    for (int j = 0; j < t; ++j)
        acc = fmaf(crow[j], Hb[(size_t)j * BU], acc);
    hprev[idx] = acc;
}

// ------------------------------------------------------------------
// Kernel 3 (per step t): rec = h_prev @ Wr (+bias1), fused GRU gates.
// Grid: 4 (b-tiles) x 16 (u-tiles) = 64 blocks of one wave.
// Each wave keeps 3 accumulators (z | r | h column blocks of Wr),
// sharing the h_prev A-fragment across the three WMMAs per K-step.
// ------------------------------------------------------------------
__global__ __launch_bounds__(32) void gru_gate_wmma_kernel(
    const float* __restrict__ xp_t,      // xp + t*B*G   : [B][768], bias0 included
    const float* __restrict__ hprev,     // [B][U]
    const float* __restrict__ Wr,        // [U][768] row-major
    const float* __restrict__ bias,      // [2][768]; row 1 = recurrent bias
    float* __restrict__ H,               // [T][B][U]
    float* __restrict__ out,             // [B][T][U]
    int t)
{
    const int lane = threadIdx.x;
    const int lr   = lane & 15;
    const int hi   = lane >> 4;

    const int tile_b = blockIdx.x & 3;       // 4 b-tiles
    const int tile_u = blockIdx.x >> 2;      // 16 u-tiles
    const int bbase  = tile_b * 16;
    const int ubase  = tile_u * 16;

    const int arow_b = bbase + lr;                       // h_prev row (batch)
    const float* arow = hprev + (size_t)arow_b * UU;

    const int nz = ubase + lr;           // z column
    const int nr = UU  + ubase + lr;     // r column (offset 256)
    const int nh = 2*UU + ubase + lr;    // h column (offset 512)

    v8f accz = {}, accr = {}, acch = {};

#pragma unroll 4
    for (int kk = 0; kk < UU; kk += 4) {
        const int k0 = kk + 2 * hi;
        v2f a = *(const v2f*)(arow + k0);            // shared A fragment

        v2f bz, br, bh;
        const float* w0 = Wr + (size_t)k0 * GG;
        const float* w1 = w0 + GG;
        bz.x = w0[nz];  bz.y = w1[nz];
        br.x = w0[nr];  br.y = w1[nr];
        bh.x = w0[nh];  bh.y = w1[nh];

        accz = __builtin_amdgcn_wmma_f32_16x16x4_f32(false, a, false, bz,
                                                     (short)0, accz, false, false);
        accr = __builtin_amdgcn_wmma_f32_16x16x4_f32(false, a, false, br,
                                                     (short)0, accr, false, false);
        acch = __builtin_amdgcn_wmma_f32_16x16x4_f32(false, a, false, bh,
                                                     (short)0, acch, false, false);
    }

    const int uu = ubase + lr;                 // output u for this lane
    const float brz = bias[GG + uu];           // bias row 1
    const float brr = bias[GG + UU + uu];
    const float brh = bias[GG + 2*UU + uu];

#pragma unroll
    for (int e = 0; e < 8; ++e) {
        const int bb = bbase + e + 8 * hi;     // batch index of this element

        const float rz = accz[e] + brz;
        const float rr = accr[e] + brr;
        const float rh = acch[e] + brh;

        const float* xrow = xp_t + (size_t)bb * GG;
        const float xz = xrow[uu];
        const float xr = xrow[UU + uu];
        const float xh = xrow[2*UU + uu];
        const float hp = hprev[(size_t)bb * UU + uu];

        const float z  = 1.0f / (1.0f + __expf(-(xz + rz)));
        const float r  = 1.0f / (1.0f + __expf(-(xr + rr)));
        const float hh = tanhf(xh + r * rh);
        const float h  = z * hp + (1.0f - z) * hh;

        H[(size_t)t * BU + (size_t)bb * UU + uu] = h;            // state for later steps
        out[((size_t)bb * TT + t) * UU + uu]     = h;            // [B,T,U] output
    }
}

// ---------------------------------------------------------------------------
extern "C" void kernel_launch(void* const* d_in, const int* in_sizes, int n_in,
                              void* d_out, int out_size, void* d_ws, size_t ws_size,
                              hipStream_t stream) {
    (void)in_sizes; (void)n_in; (void)out_size; (void)ws_size;

    const float* x    = (const float*)d_in[0];   // [B,T,D]
    const float* cond = (const float*)d_in[1];   // [B,T,T]
    const float* Wk   = (const float*)d_in[2];   // [D,3U]
    const float* Wr   = (const float*)d_in[3];   // [U,3U]
    const float* bias = (const float*)d_in[4];   // [2,3U]
    float* out = (float*)d_out;                  // [B,T,U]

    // Workspace carve-up: xp (96 MB) | H (32 MB) | hprev (64 KB)
    float* xp    = (float*)d_ws;                         // [T][B][768]
    float* H     = xp + (size_t)TT * BB * GG;            // [T][B][U]
    float* hprev = H  + (size_t)TT * BB * UU;            // [B][U]

    // Phase 1: hoisted input projection (fully parallel WMMA GEMM).
    // 2048 x 48 tiles, 4 waves/block -> 24576 blocks.
    xproj_wmma_kernel<<<24576, dim3(32, 4), 0, stream>>>(x, Wk, bias, xp);

    // Phase 2: serial DAG scan. Stream ordering carries the t -> t+1 dependency.
    for (int t = 0; t < TT; ++t) {
        hprev_kernel<<<BU / 256, 256, 0, stream>>>(cond, H, hprev, t);
        gru_gate_wmma_kernel<<<64, 32, 0, stream>>>(
            xp + (size_t)t * BG, hprev, Wr, bias, H, out, t);
    }
}